// PolicyGradientLoss_83013127897790
// MI455X (gfx1250) — compile-verified
//
#include <hip/hip_runtime.h>
#include <hip/hip_bf16.h>

#define B_DIM 256
#define S_DIM 512
#define A_DIM 2048

typedef __attribute__((ext_vector_type(2))) float v2f;
typedef __attribute__((ext_vector_type(4))) float v4f;
typedef __attribute__((ext_vector_type(8))) float v8f;

// ---------------------------------------------------------------------------
// Kernel 1: one block per (b,s) row. Stream 2048 floats once (held in regs),
// compute logsumexp + gather chosen logit, write masked chosen log-prob.
// Bandwidth-bound: 1 GiB of logits @ 23.3 TB/s ~ 46 us floor.
// ---------------------------------------------------------------------------
__global__ __launch_bounds__(256) void row_lse_kernel(
    const float* __restrict__ logits,
    const int* __restrict__ actions,
    const int* __restrict__ terminals,
    float* __restrict__ c_out) {
  const int row = blockIdx.x;              // row = b*S + s
  const int t = threadIdx.x;               // 0..255
  const int wave = t >> 5;
  const int lane = t & 31;

  const v4f* rowp = (const v4f*)(logits + (size_t)row * A_DIM);
  // Each thread owns elements [8t, 8t+8): two coalesced 128-bit loads.
  // Non-temporal: 1 GiB stream won't fit in the 192 MB L2; don't pollute it.
  v4f v0 = __builtin_nontemporal_load(&rowp[t * 2]);
  v4f v1 = __builtin_nontemporal_load(&rowp[t * 2 + 1]);
  float r[8] = {v0.x, v0.y, v0.z, v0.w, v1.x, v1.y, v1.z, v1.w};

  // ---- pass 1: row max (registers -> wave32 shuffle -> LDS cross-wave) ----
  float m = r[0];
#pragma unroll
  for (int i = 1; i < 8; ++i) m = fmaxf(m, r[i]);
#pragma unroll
  for (int off = 16; off >= 1; off >>= 1) m = fmaxf(m, __shfl_xor(m, off, 32));

  __shared__ float s_red[8];
  __shared__ float s_bcast;
  __shared__ float s_chosen;
  if (lane == 0) s_red[wave] = m;
  __syncthreads();
  if (t == 0) {
    float mm = s_red[0];
#pragma unroll
    for (int w = 1; w < 8; ++w) mm = fmaxf(mm, s_red[w]);
    s_bcast = mm;
  }
  __syncthreads();
  const float rowmax = s_bcast;

  // ---- pass 2: sum exp(x - max) from registers (no reload from memory) ----
  float sum = 0.f;
#pragma unroll
  for (int i = 0; i < 8; ++i)
    sum += exp2f((r[i] - rowmax) * 1.44269504088896340736f);  // v_exp_f32
#pragma unroll
  for (int off = 16; off >= 1; off >>= 1) sum += __shfl_xor(sum, off, 32);

  if (lane == 0) s_red[wave] = sum;         // safe: prior s_red reads were
                                            // before the last barrier
  // ---- gather chosen logit (owner thread publishes it) ----
  const int act = actions[row];
  if ((act >> 3) == t) s_chosen = r[act & 7];
  __syncthreads();

  if (t == 0) {
    float ss = 0.f;
#pragma unroll
    for (int w = 0; w < 8; ++w) ss += s_red[w];
    const float lse = rowmax + 0.69314718055994530942f * log2f(ss);
    const float mask = (terminals[row] != 0) ? 0.0f : 1.0f;
    c_out[row] = (s_chosen - lse) * mask;   // masked chosen log-prob
  }
}

// ---------------------------------------------------------------------------
// Kernel 2: one block per b. total_logp = -sum_s chosen, total_r = sum_s r,
// partial[b] = total_logp * total_r.
// ---------------------------------------------------------------------------
__global__ __launch_bounds__(256) void per_b_kernel(
    const float* __restrict__ c,
    const float* __restrict__ rewards,
    float* __restrict__ partial) {
  const int b = blockIdx.x;
  const int t = threadIdx.x;
  const int wave = t >> 5;
  const int lane = t & 31;

  const float* cb = c + (size_t)b * S_DIM;
  const float* rb = rewards + (size_t)b * S_DIM;
  float sc = cb[t] + cb[t + 256];
  float sr = rb[t] + rb[t + 256];
#pragma unroll
  for (int off = 16; off >= 1; off >>= 1) {
    sc += __shfl_xor(sc, off, 32);
    sr += __shfl_xor(sr, off, 32);
  }
  __shared__ float sC[8], sR[8];
  if (lane == 0) { sC[wave] = sc; sR[wave] = sr; }
  __syncthreads();
  if (t == 0) {
    float tc = 0.f, tr = 0.f;
#pragma unroll
    for (int w = 0; w < 8; ++w) { tc += sC[w]; tr += sR[w]; }
    partial[b] = (-tc) * tr;
  }
}

// ---------------------------------------------------------------------------
// Kernel 3: single wave32. Sum the 256 per-batch partials with
// V_WMMA_F32_16X16X4_F32: four 16x4 f32 A-tiles (64 values each) times an
// all-ones 4x16 B, accumulated into one f32 C tile. D[m][n] = rowsum_m, so
// sum of D[0..15][0] (lane 0 VGPRs + lane 16 VGPRs) = exact deterministic sum.
// f32 accumulate throughout: partials ~1e6 would overflow f16.
// EXEC is all-ones (full wave, no divergence before the WMMAs).
// ---------------------------------------------------------------------------
__global__ __launch_bounds__(32) void final_mean_kernel(
    const float* __restrict__ partial,
    float* __restrict__ out) {
  const int lane = threadIdx.x;  // 0..31
  float p[8];
#pragma unroll
  for (int i = 0; i < 8; ++i) p[i] = partial[lane * 8 + i];

  v2f ones; ones.x = 1.0f; ones.y = 1.0f;
  v8f acc = {};
#pragma unroll
  for (int i = 0; i < 4; ++i) {
    v2f a; a.x = p[2 * i]; a.y = p[2 * i + 1];
    // 8 args: (neg_a, A, neg_b, B, c_mod, C, reuse_a, reuse_b)
    acc = __builtin_amdgcn_wmma_f32_16x16x4_f32(
        false, a, false, ones, (short)0, acc, false, false);
  }
  // Column 0 of D: lane 0 holds rows 0..7, lane 16 holds rows 8..15.
  float s = acc[0] + acc[1] + acc[2] + acc[3] +
            acc[4] + acc[5] + acc[6] + acc[7];
  float s_hi = __shfl_xor(s, 16, 32);
  if (lane == 0) out[0] = (s + s_hi) * (1.0f / (float)B_DIM);
}

// ---------------------------------------------------------------------------
extern "C" void kernel_launch(void* const* d_in, const int* in_sizes, int n_in,
                              void* d_out, int out_size, void* d_ws, size_t ws_size,
                              hipStream_t stream) {
  const float* logits    = (const float*)d_in[0];  // [B,S,A] f32
  const float* rewards   = (const float*)d_in[1];  // [B,S]   f32
  const int*   actions   = (const int*)d_in[2];    // [B,S]   i32
  const int*   terminals = (const int*)d_in[3];    // [B,S]   i32
  float* out = (float*)d_out;                      // scalar f32

  float* c_ws    = (float*)d_ws;                   // B*S floats (chosen logp)
  float* partial = c_ws + (size_t)B_DIM * S_DIM;   // B floats

  row_lse_kernel<<<B_DIM * S_DIM, 256, 0, stream>>>(logits, actions, terminals, c_ws);
  per_b_kernel<<<B_DIM, 256, 0, stream>>>(c_ws, rewards, partial);
  final_mean_kernel<<<1, 32, 0, stream>>>(partial, out);
}